// CNNndpPolicy_42228118454320
// MI455X (gfx1250) — compile-verified
//
#include <hip/hip_runtime.h>
#include <cmath>

// ---- problem constants (match reference) ----
#define B_SZ     16384
#define N_BASIS  32
#define T_STEPS  300
#define DIMS     2
#define HIDN     64
#define OUTC     66                   // N*DIM + DIM
#define M_TOT    (B_SZ * DIMS)        // 32768 DMP channels
#define NTILES   19                   // ceil(300/16) step tiles -> 304 padded steps
#define DT_C     (1.0f / 300.0f)      // dt = 1/(T*L), TAU = 1

typedef float v2f __attribute__((ext_vector_type(2)));
typedef float v8f __attribute__((ext_vector_type(8)));

// Packed operand layouts (WMMA f32 16x16x4, wave32):
//   lane l: row/col = l&15, K sub-pair offset = 2*(l>>4)
//   element i = 2*kk + e  <->  K index n = 4*kk + 2*(l>>4) + e
// B feed:  phiP[(j*32 + l)*16 + i]   (j = step tile)     -> 64B/lane contiguous
// A feed:  WgP [(mgroup*32 + l)*16 + i] (mgroup = m/16)  -> 64B/lane contiguous

// ---------------------------------------------------------------------------
// Kernel 1: batch-independent phase/RBF table, written in packed B-feed order.
// phi_raw[t][n] = x_{t+1} * psi_n(x_{t+1}) / sum_n psi_n,  x_s = (1-dt)^s.
// Rows 300..303 zero-padded for the 19th step tile.
// ---------------------------------------------------------------------------
__global__ void phase_kernel(const float* __restrict__ c,
                             const float* __restrict__ sig2,
                             float* __restrict__ phiP) {
  int t = blockIdx.x * blockDim.x + threadIdx.x;
  if (t >= NTILES * 16) return;

  float vals[N_BASIS];
  if (t >= T_STEPS) {
    for (int n = 0; n < N_BASIS; ++n) vals[n] = 0.0f;
  } else {
    float x = powf(1.0f - DT_C, (float)(t + 1));
    float s = 0.0f;
    for (int n = 0; n < N_BASIS; ++n) {
      float d = x - c[n];
      float p = expf(-0.5f * d * d / sig2[n]);
      vals[n] = p;
      s += p;
    }
    float scale = x / s;
    for (int n = 0; n < N_BASIS; ++n) vals[n] *= scale;
  }

  int j = t >> 4, colm = t & 15;
  for (int n = 0; n < N_BASIS; ++n) {
    int kk = n >> 2, half = (n >> 1) & 1, e = n & 1;
    int l = colm + (half << 4);
    phiP[((j * 32 + l) << 4) + (kk << 1) + e] = vals[n];
  }
}

// ---------------------------------------------------------------------------
// Kernel 2: per-row MLP (one thread per batch row; weights lane-uniform ->
// scalar-cached loads). Emits packed W, k=goal-y0, y0, a_z, v, sigma.
// ---------------------------------------------------------------------------
__global__ void mlp_kernel(const float* __restrict__ state,
                           const float* __restrict__ w1, const float* __restrict__ b1,
                           const float* __restrict__ w2, const float* __restrict__ b2,
                           const float* __restrict__ sw, const float* __restrict__ sb,
                           const float* __restrict__ vw, const float* __restrict__ vb,
                           float* __restrict__ WgP, float* __restrict__ Kc,
                           float* __restrict__ Y0, float* __restrict__ Az,
                           float* __restrict__ Vv, float* __restrict__ out_sigma) {
  int b = blockIdx.x * blockDim.x + threadIdx.x;
  if (b >= B_SZ) return;
  float s0 = state[b * 3 + 0], s1 = state[b * 3 + 1], s2 = state[b * 3 + 2];

  float h[HIDN];
#pragma unroll
  for (int j = 0; j < HIDN; ++j) {
    float a = w1[j * 3] * s0 + w1[j * 3 + 1] * s1 + w1[j * 3 + 2] * s2 + b1[j];
    h[j] = tanhf(a) * 0.1f;
  }

  for (int o = 0; o < OUTC; ++o) {
    float acc = b2[o];
#pragma unroll
    for (int j = 0; j < HIDN; ++j) acc += h[j] * w2[o * HIDN + j];
    if (o < DIMS) {
      int m = b * DIMS + o;
      float y0v = (o == 0) ? s0 : s1;        // y0 = state[:, :2]
      Y0[m] = y0v;
      Kc[m] = acc - y0v;                      // goal - y0 (forcing scale)
    } else {
      int q = o - DIMS;                       // ndp_wg[:,2:66].reshape(-1,32)
      int m = b * DIMS + (q >> 5);
      int n = q & 31;
      int l = (m & 15) + (((n >> 1) & 1) << 4);
      WgP[((m >> 4) << 9) + (l << 4) + ((n >> 2) << 1) + (n & 1)] = acc;
      if (o == OUTC - 1)                      // last col doubles as a_z
        Az[b] = fminf(fmaxf(acc, 0.5f), 30.0f);
    }
  }

  for (int d = 0; d < DIMS; ++d) {
    float as = sb[d], av = vb[d];
#pragma unroll
    for (int j = 0; j < HIDN; ++j) {
      as += h[j] * sw[d * HIDN + j];
      av += h[j] * vw[d * HIDN + j];
    }
    out_sigma[b * DIMS + d] = 1.0f / (1.0f + expf(-as)) + 0.001f;
    Vv[b * DIMS + d] = av;
  }
}

// ---------------------------------------------------------------------------
// Kernel 3: fused  F = W @ Phi^T  (fp32 WMMA)  ->  LDS  ->  Euler scan  ->
// coalesced stores of a = diff(Y).
// Per wave: 16 channels, 19 step tiles x 8 V_WMMA_F32_16X16X4_F32.
// B operand double-buffered across tiles so global-load latency overlaps the
// 8 chained WMMAs + LDS dump of the previous tile.
// ---------------------------------------------------------------------------
__global__ __launch_bounds__(64)
void dmp_kernel(const float* __restrict__ WgP, const float* __restrict__ phiP,
                const float* __restrict__ Kc, const float* __restrict__ Y0,
                const float* __restrict__ Az, float* __restrict__ out_a) {
  __shared__ float Flds[2 * 16 * 305];   // 2 waves * 16 ch * 305 (odd stride)

  const int lane   = threadIdx.x & 31;
  const int wave   = threadIdx.x >> 5;
  const int colm   = lane & 15;
  const int mbase  = blockIdx.x * 32 + wave * 16;
  const int mgroup = (blockIdx.x << 1) | wave;
  float* fw = &Flds[wave * 16 * 305];

  const float* abase = WgP + ((mgroup << 5) + lane) * 16;  // 64B/lane, aligned
  const float* bbase = phiP + (lane << 4);                 // + j*512 per tile

  __builtin_prefetch(abase, 0, 3);       // -> global_prefetch_b8
  __builtin_prefetch(bbase, 0, 3);

  v2f A[8];
#pragma unroll
  for (int kk = 0; kk < 8; ++kk) A[kk] = *(const v2f*)(abase + 2 * kk);

  v2f B0[8], B1[8];
#pragma unroll
  for (int kk = 0; kk < 8; ++kk) B0[kk] = *(const v2f*)(bbase + 2 * kk);

  const int rbase = (lane >> 4) * 8;
#pragma unroll
  for (int j = 0; j < NTILES; ++j) {
    if (j + 1 < NTILES) {                // prefetch next tile's B into regs
      const float* nb = bbase + (j + 1) * 512;
#pragma unroll
      for (int kk = 0; kk < 8; ++kk) B1[kk] = *(const v2f*)(nb + 2 * kk);
    }
    v8f acc = {0.f, 0.f, 0.f, 0.f, 0.f, 0.f, 0.f, 0.f};
#pragma unroll
    for (int kk = 0; kk < 8; ++kk)
      acc = __builtin_amdgcn_wmma_f32_16x16x4_f32(
          false, A[kk], false, B0[kk], (short)0, acc, false, false);
#pragma unroll
    for (int r = 0; r < 8; ++r)
      fw[(rbase + r) * 305 + j * 16 + colm] = acc[r];
#pragma unroll
    for (int kk = 0; kk < 8; ++kk) B0[kk] = B1[kk];
  }

  asm volatile("s_wait_dscnt 0" ::: "memory");

  // Sequential Euler scan: lane l owns channel mbase+l (l < 16).
  // a[t] = Y[t+1]-Y[t] = dt * z (pre-update z); overwrite F in place.
  if (lane < 16) {
    int m    = mbase + lane;
    float k  = Kc[m];
    float y  = Y0[m];
    float g  = k + y;
    float z  = 0.01f;                 // z0 = dy0 * tau
    float az = Az[m >> 1];
    float bz = az * 0.25f;
    float* frow = &fw[lane * 305];
    for (int t = 0; t < T_STEPS; ++t) {
      float fx = frow[t] * k;
      float dz = az * (bz * (g - y) - z) + fx;
      frow[t]  = DT_C * z;            // a output
      y += DT_C * z;
      z += DT_C * dz;
    }
  }

  asm volatile("s_wait_dscnt 0" ::: "memory");

  // Coalesced write-out: full wave streams each channel's 300 floats.
  for (int ch = 0; ch < 16; ++ch) {
    int m = mbase + ch;
    for (int t = lane; t < T_STEPS; t += 32)
      out_a[m * T_STEPS + t] = fw[ch * 305 + t];
  }
}

// ---------------------------------------------------------------------------
// Kernel 4: value broadcast (fully coalesced; v[m] repeated 300x).
// ---------------------------------------------------------------------------
__global__ void value_kernel(const float* __restrict__ Vv,
                             float* __restrict__ out_val) {
  int i = blockIdx.x * blockDim.x + threadIdx.x;
  const int total = M_TOT * T_STEPS;
  if (i < total) out_val[i] = Vv[i / T_STEPS];
}

// ---------------------------------------------------------------------------
extern "C" void kernel_launch(void* const* d_in, const int* in_sizes, int n_in,
                              void* d_out, int out_size, void* d_ws, size_t ws_size,
                              hipStream_t stream) {
  const float* state  = (const float*)d_in[0];
  const float* fc1_w  = (const float*)d_in[1];
  const float* fc1_b  = (const float*)d_in[2];
  const float* fc2m_w = (const float*)d_in[3];
  const float* fc2m_b = (const float*)d_in[4];
  const float* sig_w  = (const float*)d_in[5];
  const float* sig_b  = (const float*)d_in[6];
  const float* val_w  = (const float*)d_in[7];
  const float* val_b  = (const float*)d_in[8];
  const float* dmp_c  = (const float*)d_in[9];
  const float* dmp_s2 = (const float*)d_in[10];

  float* out = (float*)d_out;
  float* ws  = (float*)d_ws;

  // workspace layout (floats): ~4.7 MB total
  float* phiP = ws;                       // 19*32*16 = 9728 (reserve 10240)
  float* WgP  = ws + 10240;               // 32768*32 packed
  float* Kc   = WgP + (size_t)M_TOT * N_BASIS;
  float* Y0   = Kc + M_TOT;
  float* Az   = Y0 + M_TOT;
  float* Vv   = Az + B_SZ;

  // output layout: a | sigma | value
  float* out_a   = out;
  float* out_sig = out + (size_t)M_TOT * T_STEPS;
  float* out_val = out_sig + M_TOT;

  phase_kernel<<<1, 320, 0, stream>>>(dmp_c, dmp_s2, phiP);
  mlp_kernel<<<B_SZ / 256, 256, 0, stream>>>(state, fc1_w, fc1_b, fc2m_w, fc2m_b,
                                             sig_w, sig_b, val_w, val_b,
                                             WgP, Kc, Y0, Az, Vv, out_sig);
  dmp_kernel<<<M_TOT / 32, 64, 0, stream>>>(WgP, phiP, Kc, Y0, Az, out_a);

  const int total = M_TOT * T_STEPS;
  value_kernel<<<(total + 255) / 256, 256, 0, stream>>>(Vv, out_val);
}